// Net_19009525252327
// MI455X (gfx1250) — compile-verified
//
#include <hip/hip_runtime.h>
#include <hip/hip_bf16.h>

typedef float v2f __attribute__((ext_vector_type(2)));
typedef float v8f __attribute__((ext_vector_type(8)));

#define F_IN 35
#define F_H  64
#define F_OUT 2

// ---------------------------------------------------------------------------
// Degree: deg[n] = 1 (self loop) + sum(ew | dst==n)
// ---------------------------------------------------------------------------
__global__ void k_init_deg(float* __restrict__ deg, int n) {
    int i = blockIdx.x * blockDim.x + threadIdx.x;
    if (i < n) deg[i] = 1.0f;
}

__global__ void k_accum_deg(const long long* __restrict__ dst,
                            const float* __restrict__ ew,
                            float* __restrict__ deg, int e) {
    int i = blockIdx.x * blockDim.x + threadIdx.x;
    if (i < e) atomicAdd(&deg[(int)dst[i]], ew[i]);
}

__global__ void k_finish_dinv(float* __restrict__ deg, int n) {
    int i = blockIdx.x * blockDim.x + threadIdx.x;
    if (i < n) {
        float d = deg[i];
        deg[i] = (d > 0.0f) ? rsqrtf(d) : 0.0f;
    }
}

// ---------------------------------------------------------------------------
// Layer-1 GEMM: h1[N,64] = x[N,35] @ W1[35,64], fp32 WMMA 16x16x4.
// Block = 128 threads = 4 waves; wave w computes tile rows [16*bid, +16),
// cols [16*w, +16). K padded 35 -> 36 (9 steps of 4).
// A frag (16x4): lane l, vgpr v -> A[l%16][k0 + v + (l>>4)*2]
// B frag (4x16): lane l, vgpr v -> B[k0 + v + (l>>4)*2][l%16 of tile]
// D (16x16):     lane l, vgpr j -> D[j + (l>>4)*8][l%16 of tile]
// ---------------------------------------------------------------------------
__global__ void k_gemm1_wmma(const float* __restrict__ x,
                             const float* __restrict__ W1,
                             float* __restrict__ h1, int n) {
    const int lane  = threadIdx.x & 31;
    const int wave  = threadIdx.x >> 5;        // 0..3 -> column tile
    const int rbase = blockIdx.x * 16;
    const int cbase = wave * 16;
    const int m     = lane & 15;
    const int khalf = (lane >> 4) * 2;         // 0 or 2

    int row = rbase + m;
    if (row >= n) row = n - 1;                 // clamp (select, keeps EXEC full)
    const float* xrow = x + (size_t)row * F_IN;
    const int col = cbase + m;

    v8f acc = {};
    #pragma unroll
    for (int k0 = 0; k0 < 36; k0 += 4) {
        const int ka0 = k0 + khalf;
        const int ka1 = ka0 + 1;
        v2f a, b;
        a.x = (ka0 < F_IN) ? xrow[ka0] : 0.0f;
        a.y = (ka1 < F_IN) ? xrow[ka1] : 0.0f;
        b.x = (ka0 < F_IN) ? W1[ka0 * F_H + col] : 0.0f;
        b.y = (ka1 < F_IN) ? W1[ka1 * F_H + col] : 0.0f;
        acc = __builtin_amdgcn_wmma_f32_16x16x4_f32(
            /*neg_a=*/false, a, /*neg_b=*/false, b,
            /*c_mod=*/(short)0, acc, /*reuse_a=*/false, /*reuse_b=*/false);
    }

    const int row0 = rbase + (lane >> 4) * 8;
    #pragma unroll
    for (int j = 0; j < 8; ++j) {
        int r = row0 + j;
        if (r < n) h1[(size_t)r * F_H + col] = acc[j];
    }
}

// ---------------------------------------------------------------------------
// out1[n][f] = b1[f] + dinv[n]^2 * h1[n][f]   (bias + self-loop message)
// ---------------------------------------------------------------------------
__global__ void k_init_out1(float* __restrict__ out1,
                            const float* __restrict__ h1,
                            const float* __restrict__ b1,
                            const float* __restrict__ dinv, int n) {
    long long idx = (long long)blockIdx.x * blockDim.x + threadIdx.x;
    if (idx >= (long long)n * F_H) return;
    int node = (int)(idx >> 6);
    int f    = (int)(idx & 63);
    float di = dinv[node];
    out1[idx] = b1[f] + di * di * h1[idx];
}

// ---------------------------------------------------------------------------
// Layer-1 message passing: one wave per edge; lane handles 2 features.
// out1[dst] += dinv[src]*ew*dinv[dst] * h1[src]
// ---------------------------------------------------------------------------
__global__ void k_mp1(const long long* __restrict__ src,
                      const long long* __restrict__ dst,
                      const float* __restrict__ ew,
                      const float* __restrict__ dinv,
                      const float* __restrict__ h1,
                      float* __restrict__ out1, int e) {
    long long t    = (long long)blockIdx.x * blockDim.x + threadIdx.x;
    long long edge = t >> 5;
    if (edge >= e) return;
    const int lane = (int)(t & 31);
    const int s = (int)src[edge];
    const int d = (int)dst[edge];
    const float norm = dinv[s] * ew[edge] * dinv[d];
    const float2 hv = *(const float2*)(h1 + (size_t)s * F_H + lane * 2);
    float* o = out1 + (size_t)d * F_H + lane * 2;
    atomicAdd(o,     norm * hv.x);
    atomicAdd(o + 1, norm * hv.y);
}

// ---------------------------------------------------------------------------
// Layer-2 GEMM fused with ReLU: h2[n][c] = sum_k relu(out1[n][k]) * W2[k][c]
// (64->2 is too skinny for WMMA; scalar dot is optimal)
// ---------------------------------------------------------------------------
__global__ void k_layer2(const float* __restrict__ out1,
                         const float* __restrict__ W2,
                         float* __restrict__ h2, int n) {
    int i = blockIdx.x * blockDim.x + threadIdx.x;
    if (i >= n) return;
    const float* r = out1 + (size_t)i * F_H;
    float a0 = 0.0f, a1 = 0.0f;
    #pragma unroll
    for (int k = 0; k < F_H; ++k) {
        float v = fmaxf(r[k], 0.0f);
        a0 = fmaf(v, W2[2 * k],     a0);
        a1 = fmaf(v, W2[2 * k + 1], a1);
    }
    h2[2 * i]     = a0;
    h2[2 * i + 1] = a1;
}

// out[n][c] = b2[c] + dinv[n]^2 * h2[n][c]
__global__ void k_init_out2(float* __restrict__ out,
                            const float* __restrict__ h2,
                            const float* __restrict__ b2,
                            const float* __restrict__ dinv, int n) {
    int idx = blockIdx.x * blockDim.x + threadIdx.x;
    if (idx >= n * F_OUT) return;
    int node = idx >> 1;
    int c    = idx & 1;
    float di = dinv[node];
    out[idx] = b2[c] + di * di * h2[idx];
}

// Layer-2 message passing: one thread per edge (2 features).
__global__ void k_mp2(const long long* __restrict__ src,
                      const long long* __restrict__ dst,
                      const float* __restrict__ ew,
                      const float* __restrict__ dinv,
                      const float* __restrict__ h2,
                      float* __restrict__ out, int e) {
    int i = blockIdx.x * blockDim.x + threadIdx.x;
    if (i >= e) return;
    const int s = (int)src[i];
    const int d = (int)dst[i];
    const float norm = dinv[s] * ew[i] * dinv[d];
    atomicAdd(&out[2 * d],     norm * h2[2 * s]);
    atomicAdd(&out[2 * d + 1], norm * h2[2 * s + 1]);
}

// In-place log_softmax over the 2 classes.
__global__ void k_logsoftmax(float* __restrict__ out, int n) {
    int i = blockIdx.x * blockDim.x + threadIdx.x;
    if (i >= n) return;
    float z0 = out[2 * i];
    float z1 = out[2 * i + 1];
    float m  = fmaxf(z0, z1);
    float lse = m + logf(expf(z0 - m) + expf(z1 - m));
    out[2 * i]     = z0 - lse;
    out[2 * i + 1] = z1 - lse;
}

// ---------------------------------------------------------------------------
extern "C" void kernel_launch(void* const* d_in, const int* in_sizes, int n_in,
                              void* d_out, int out_size, void* d_ws, size_t ws_size,
                              hipStream_t stream) {
    const float*     x  = (const float*)d_in[0];
    const long long* ei = (const long long*)d_in[1];   // int64 [2, E]
    const float*     ew = (const float*)d_in[2];
    const float*     W1 = (const float*)d_in[3];
    const float*     b1 = (const float*)d_in[4];
    const float*     W2 = (const float*)d_in[5];
    const float*     b2 = (const float*)d_in[6];
    float* out = (float*)d_out;

    const int n = in_sizes[0] / F_IN;   // 100000
    const int e = in_sizes[2];          // 3200000
    const long long* src = ei;
    const long long* dst = ei + e;

    // Workspace layout (floats): dinv[n] | h1[n*64] | out1[n*64] | h2[n*2]
    float* ws   = (float*)d_ws;
    float* dinv = ws;
    float* h1   = ws + n;
    float* out1 = ws + n + (size_t)n * F_H;
    float* h2   = ws + n + (size_t)n * F_H * 2;

    const int B = 256;

    // 1) weighted in-degree (self-loop folded in as +1) -> dinv = rsqrt(deg)
    k_init_deg  <<<(n + B - 1) / B, B, 0, stream>>>(dinv, n);
    k_accum_deg <<<(e + B - 1) / B, B, 0, stream>>>(dst, ew, dinv, e);
    k_finish_dinv<<<(n + B - 1) / B, B, 0, stream>>>(dinv, n);

    // 2) h1 = x @ W1 via fp32 WMMA (4 column tiles per block of 4 waves)
    k_gemm1_wmma<<<(n + 15) / 16, 128, 0, stream>>>(x, W1, h1, n);

    // 3) out1 = b1 + self-loop message
    long long tot1 = (long long)n * F_H;
    k_init_out1<<<(int)((tot1 + B - 1) / B), B, 0, stream>>>(out1, h1, b1, dinv, n);

    // 4) edge scatter layer 1 (wave per edge)
    long long tmp = (long long)e * 32;
    k_mp1<<<(int)((tmp + B - 1) / B), B, 0, stream>>>(src, dst, ew, dinv, h1, out1, e);

    // 5) h2 = relu(out1) @ W2
    k_layer2<<<(n + B - 1) / B, B, 0, stream>>>(out1, W2, h2, n);

    // 6) out = b2 + self-loop message (directly into d_out)
    k_init_out2<<<(n * F_OUT + B - 1) / B, B, 0, stream>>>(out, h2, b2, dinv, n);

    // 7) edge scatter layer 2
    k_mp2<<<(e + B - 1) / B, B, 0, stream>>>(src, dst, ew, dinv, h2, out, e);

    // 8) log_softmax in place
    k_logsoftmax<<<(n + B - 1) / B, B, 0, stream>>>(out, n);
}